// Net_23356032155768
// MI455X (gfx1250) — compile-verified
//
#include <hip/hip_runtime.h>
#include <hip/hip_bf16.h>

typedef float v2f __attribute__((ext_vector_type(2)));
typedef float v8f __attribute__((ext_vector_type(8)));

#define N_NODES 100000
#define N_EDGES 1600000
#define F_IN    256
#define HID     64
#define NCLS    16

// ---------------------------------------------------------------------------
// zero-fill scratch (accumulators must be re-zeroed every launch)
// ---------------------------------------------------------------------------
__global__ void zero_f32(float* p, long long n) {
    long long i = (long long)blockIdx.x * blockDim.x + threadIdx.x;
    long long stride = (long long)gridDim.x * blockDim.x;
    for (; i < n; i += stride) p[i] = 0.0f;
}

// ---------------------------------------------------------------------------
// Dense GEMM with fp32 WMMA: H[nrows,NCOL] = X[nrows,K] @ W[K,NCOL] + bias
// One wave computes a 16-row x NCOL tile; CT = NCOL/16 accumulator tiles.
// fp32 A(16x4) fragment: lanes 0-15 hold K={k,k+1}, lanes 16-31 K={k+2,k+3},
// row M = lane%16.  B(4x16): same K split, col N = lane%16.
// C/D(16x16): VGPR j -> row j + 8*(lane/16), col = lane%16.   (ISA 7.12.2)
// ---------------------------------------------------------------------------
template <int CT, int K, int NCOL>
__global__ void gemm_wmma_f32(const float* __restrict__ X,
                              const float* __restrict__ W,
                              const float* __restrict__ bias,
                              float* __restrict__ H,
                              int nrows) {
    const int lane  = threadIdx.x & 31;
    const int wib   = threadIdx.x >> 5;
    const int tile  = blockIdx.x * (blockDim.x >> 5) + wib;   // 16-row tile
    const int m0    = tile * 16;
    if (m0 >= nrows) return;            // wave-uniform: EXEC stays all-ones

    const int half = lane >> 4;         // 0 | 1
    const int l16  = lane & 15;

    v8f acc[CT] = {};                   // zero-initialized accumulators

    const float* __restrict__ xrow = X + (long long)(m0 + l16) * K;

    #pragma unroll 4
    for (int k = 0; k < K; k += 4) {
        const int ka = k + 2 * half;    // this lane's K pair
        v2f a;
        a.x = xrow[ka];
        a.y = xrow[ka + 1];
        #pragma unroll
        for (int t = 0; t < CT; ++t) {
            const int col = t * 16 + l16;
            v2f b;
            b.x = W[(long long)ka * NCOL + col];
            b.y = W[(long long)(ka + 1) * NCOL + col];
            acc[t] = __builtin_amdgcn_wmma_f32_16x16x4_f32(
                false, a, false, b, (short)0, acc[t], false, false);
        }
    }

    #pragma unroll
    for (int t = 0; t < CT; ++t) {
        const int col = t * 16 + l16;
        const float bv = bias[col];
        #pragma unroll
        for (int j = 0; j < 8; ++j) {
            const int m = m0 + j + 8 * half;
            H[(long long)m * NCOL + col] = acc[t][j] + bv;
        }
    }
}

// ---------------------------------------------------------------------------
// per-node attention scores: s_src = h . a[:d], s_dst = h . a[d:]
// ---------------------------------------------------------------------------
__global__ void node_scores(const float* __restrict__ h,
                            const float* __restrict__ a,
                            float* __restrict__ ssrc,
                            float* __restrict__ sdst,
                            int n, int d) {
    int i = blockIdx.x * blockDim.x + threadIdx.x;
    if (i >= n) return;
    const float* r = h + (long long)i * d;
    float s0 = 0.0f, s1 = 0.0f;
    for (int j = 0; j < d; ++j) {
        float v = r[j];
        s0 += v * a[j];
        s1 += v * a[d + j];
    }
    ssrc[i] = s0;
    sdst[i] = s1;
}

// ---------------------------------------------------------------------------
// edge weights: e = lrelu(s_src[row]+s_dst[col]); w = exp(e);
// denom[col] += w  (softmax denominator via atomics, matches segment_sum)
// ---------------------------------------------------------------------------
__global__ void edge_scores(const int* __restrict__ ei,
                            const float* __restrict__ ssrc,
                            const float* __restrict__ sdst,
                            float* __restrict__ w,
                            float* __restrict__ denom,
                            int E, int T) {
    int i = blockIdx.x * blockDim.x + threadIdx.x;
    if (i >= T) return;
    int r = (i < E) ? ei[i]     : (i - E);   // source (self-loop past E)
    int c = (i < E) ? ei[E + i] : (i - E);   // target
    float e = ssrc[r] + sdst[c];
    e = (e > 0.0f) ? e : 0.2f * e;           // LeakyReLU(0.2)
    float ex = __expf(e);                    // unstabilized exp, per reference
    w[i] = ex;
    atomicAdd(&denom[c], ex);
}

// ---------------------------------------------------------------------------
// layer-1 aggregation: out[col] += (w/denom[col]) * h[row], 64 features.
// One wave per edge; each lane covers 2 columns.
// ---------------------------------------------------------------------------
__global__ void aggregate64(const int* __restrict__ ei,
                            const float* __restrict__ h,
                            const float* __restrict__ w,
                            const float* __restrict__ denom,
                            float* __restrict__ out,
                            int E, int T) {
    int wave = (int)((blockIdx.x * blockDim.x + threadIdx.x) >> 5);
    int lane = threadIdx.x & 31;
    if (wave >= T) return;
    int r = (wave < E) ? ei[wave]     : (wave - E);
    int c = (wave < E) ? ei[E + wave] : (wave - E);
    float coef = w[wave] / denom[c];
    const float* hr = h + (long long)r * 64;
    float* oc = out + (long long)c * 64;
    atomicAdd(&oc[lane],      coef * hr[lane]);
    atomicAdd(&oc[lane + 32], coef * hr[lane + 32]);
}

// ---------------------------------------------------------------------------
// layer-2 aggregation: 16 features; thread -> (edge, col)
// ---------------------------------------------------------------------------
__global__ void aggregate16(const int* __restrict__ ei,
                            const float* __restrict__ h,
                            const float* __restrict__ w,
                            const float* __restrict__ denom,
                            float* __restrict__ out,
                            int E, int T) {
    long long idx = (long long)blockIdx.x * blockDim.x + threadIdx.x;
    int e   = (int)(idx >> 4);
    int col = (int)(idx & 15);
    if (e >= T) return;
    int r = (e < E) ? ei[e]     : (e - E);
    int c = (e < E) ? ei[E + e] : (e - E);
    float coef = w[e] / denom[c];
    atomicAdd(&out[(long long)c * 16 + col], coef * h[(long long)r * 16 + col]);
}

// ---------------------------------------------------------------------------
// row-wise log_softmax over 16 classes
// ---------------------------------------------------------------------------
__global__ void log_softmax16(const float* __restrict__ in,
                              float* __restrict__ out, int n) {
    int i = blockIdx.x * blockDim.x + threadIdx.x;
    if (i >= n) return;
    const float* r = in + (long long)i * 16;
    float v[16];
    float m = r[0];
    #pragma unroll
    for (int j = 0; j < 16; ++j) { v[j] = r[j]; m = fmaxf(m, v[j]); }
    float s = 0.0f;
    #pragma unroll
    for (int j = 0; j < 16; ++j) { v[j] -= m; s += __expf(v[j]); }
    float ls = __logf(s);
    float* o = out + (long long)i * 16;
    #pragma unroll
    for (int j = 0; j < 16; ++j) o[j] = v[j] - ls;
}

// ---------------------------------------------------------------------------
// orchestration
// ---------------------------------------------------------------------------
extern "C" void kernel_launch(void* const* d_in, const int* in_sizes, int n_in,
                              void* d_out, int out_size, void* d_ws, size_t ws_size,
                              hipStream_t stream) {
    const float* x   = (const float*)d_in[0];
    const int*   ei  = (const int*)  d_in[1];   // [2,E] int32
    const float* W1  = (const float*)d_in[2];
    const float* b1  = (const float*)d_in[3];
    const float* a1  = (const float*)d_in[4];
    const float* W2  = (const float*)d_in[5];
    const float* b2  = (const float*)d_in[6];
    const float* a2  = (const float*)d_in[7];
    float* out = (float*)d_out;

    const int N = in_sizes[0] / F_IN;      // 100000
    const int E = in_sizes[1] / 2;         // 1600000
    const int T = E + N;                   // edges incl. self-loops

    // workspace carve-up (floats)
    float* ws    = (float*)d_ws;
    float* h1    = ws;                                 // N*64
    float* out1  = h1   + (long long)N * HID;          // N*64
    float* h2    = out1 + (long long)N * HID;          // N*16
    float* out2  = h2   + (long long)N * NCLS;         // N*16
    float* ssrc  = out2 + (long long)N * NCLS;         // N
    float* sdst  = ssrc + N;                           // N
    float* denom = sdst + N;                           // N
    float* wbuf  = denom + N;                          // T

    const int ZB = 1024;
    const int tilesPerBlk = 8;                         // 256 thr = 8 waves
    const int gemmBlocks  = (N / 16 + tilesPerBlk - 1) / tilesPerBlk;

    // ---------------- layer 1 ----------------
    zero_f32<<<2048, 256, 0, stream>>>(out1, (long long)N * HID);
    zero_f32<<<ZB, 256, 0, stream>>>(denom, (long long)N);

    gemm_wmma_f32<HID / 16, F_IN, HID>
        <<<gemmBlocks, 256, 0, stream>>>(x, W1, b1, h1, N);

    node_scores<<<(N + 255) / 256, 256, 0, stream>>>(h1, a1, ssrc, sdst, N, HID);

    edge_scores<<<(T + 255) / 256, 256, 0, stream>>>(ei, ssrc, sdst, wbuf, denom, E, T);

    aggregate64<<<(T + 7) / 8, 256, 0, stream>>>(ei, h1, wbuf, denom, out1, E, T);

    // ---------------- layer 2 ----------------
    zero_f32<<<ZB, 256, 0, stream>>>(out2, (long long)N * NCLS);
    zero_f32<<<ZB, 256, 0, stream>>>(denom, (long long)N);

    gemm_wmma_f32<NCLS / 16, HID, NCLS>
        <<<gemmBlocks, 256, 0, stream>>>(out1, W2, b2, h2, N);

    node_scores<<<(N + 255) / 256, 256, 0, stream>>>(h2, a2, ssrc, sdst, N, NCLS);

    edge_scores<<<(T + 255) / 256, 256, 0, stream>>>(ei, ssrc, sdst, wbuf, denom, E, T);

    {
        long long work = (long long)T * NCLS;
        int blocks = (int)((work + 255) / 256);
        aggregate16<<<blocks, 256, 0, stream>>>(ei, h2, wbuf, denom, out2, E, T);
    }

    log_softmax16<<<(N + 255) / 256, 256, 0, stream>>>(out2, out, N);
}